// GCN_5995774345336
// MI455X (gfx1250) — compile-verified
//
#include <hip/hip_runtime.h>
#include <cstdint>

#define NN 100000
#define NE 640000
#define DI 128
#define DH 128
#define DO 64

typedef __attribute__((ext_vector_type(2))) float v2f;
typedef __attribute__((ext_vector_type(4))) float v4f;
typedef __attribute__((ext_vector_type(8))) float v8f;

__global__ __launch_bounds__(256) void k_degree(const int* __restrict__ dst,
                                                float* __restrict__ deg, int E) {
  int e = blockIdx.x * blockDim.x + threadIdx.x;
  if (e < E) atomicAdd(&deg[dst[e]], 1.0f);
}

__global__ __launch_bounds__(256) void k_dinv(float* __restrict__ deg, int n) {
  int i = blockIdx.x * blockDim.x + threadIdx.x;
  if (i < n) deg[i] = rsqrtf(deg[i] + 1.0f);  // +1 = self loop
}

__global__ __launch_bounds__(256) void k_norm(const int* __restrict__ src,
                                              const int* __restrict__ dst,
                                              const float* __restrict__ dinv,
                                              float* __restrict__ norm, int E) {
  int e = blockIdx.x * blockDim.x + threadIdx.x;
  if (e < E) norm[e] = dinv[src[e]] * dinv[dst[e]];
}

// H = X @ W.  X:[rows,DI] row-major, W:[DI, COLT*16] row-major, rows % 16 == 0.
// One wave32 computes a 16x16 output tile via V_WMMA_F32_16X16X4_F32 (K loop of 32).
template <int COLT>
__global__ __launch_bounds__(COLT * 32) void k_gemm_wmma(
    const float* __restrict__ X, const float* __restrict__ W,
    float* __restrict__ H) {
  constexpr int DOUT = COLT * 16;
  const int lane = threadIdx.x & 31;
  const int wave = threadIdx.x >> 5;
  const int row0 = blockIdx.x * 16;
  const int col0 = wave * 16;
  const int sel = lane & 15;             // A: row in tile, B/C: col in tile
  const int khalf = (lane >> 4) << 1;    // 0 (lanes 0-15) or 2 (lanes 16-31)

  const float* Arow = X + (size_t)(row0 + sel) * DI + khalf;
  const float* Bcol = W + (size_t)khalf * DOUT + col0 + sel;

  v8f acc = {};
#pragma unroll
  for (int k0 = 0; k0 < DI; k0 += 4) {
    v2f a = *(const v2f*)(Arow + k0);              // A[row][k0+khalf .. +1]
    v2f b;
    b.x = Bcol[(size_t)k0 * DOUT];                 // B[k0+khalf][col]
    b.y = Bcol[(size_t)(k0 + 1) * DOUT];           // B[k0+khalf+1][col]
    acc = __builtin_amdgcn_wmma_f32_16x16x4_f32(false, a, false, b, (short)0,
                                                acc, false, false);
  }
  // C/D layout: VGPR r, lane l -> row = r + 8*(l>=16), col = l%16
  const int mbase = row0 + ((lane >> 4) << 3);
  float* out = H + col0 + sel;
#pragma unroll
  for (int r = 0; r < 8; ++r) out[(size_t)(mbase + r) * DOUT] = acc[r];
}

// agg[dst] += h[src] * norm  (edge-parallel, float4 per thread, f32 atomics)
template <int D>
__global__ __launch_bounds__(256) void k_scatter(
    const float* __restrict__ H, const int* __restrict__ src,
    const int* __restrict__ dst, const float* __restrict__ norm,
    float* __restrict__ agg, int E) {
  constexpr int CH = D / 4;
  unsigned t = blockIdx.x * 256u + threadIdx.x;
  if (t >= (unsigned)E * CH) return;
  int e = t / CH;
  int c = t % CH;
  int s = src[e];
  int d = dst[e];
  float w = norm[e];
  v4f h = *(const v4f*)(H + (size_t)s * D + c * 4);
  float* a = agg + (size_t)d * D + c * 4;
  atomicAdd(a + 0, h.x * w);
  atomicAdd(a + 1, h.y * w);
  atomicAdd(a + 2, h.z * w);
  atomicAdd(a + 3, h.w * w);
}

// out = maybe_relu(agg + h*dinv[i]^2 + bias)   (self-loop folded in, no atomics)
template <int D, bool RELU>
__global__ __launch_bounds__(256) void k_finish(
    const float* __restrict__ agg, const float* __restrict__ H,
    const float* __restrict__ dinv, const float* __restrict__ bias,
    float* __restrict__ out, int n) {
  constexpr int CH = D / 4;
  unsigned t = blockIdx.x * 256u + threadIdx.x;
  if (t >= (unsigned)n * CH) return;
  int i = t / CH;
  int c = t % CH;
  float sn = dinv[i];
  sn *= sn;
  v4f a = *(const v4f*)(agg + (size_t)i * D + c * 4);
  v4f h = *(const v4f*)(H + (size_t)i * D + c * 4);
  v4f b = *(const v4f*)(bias + c * 4);
  v4f r = a + h * sn + b;
  if (RELU) {
    r.x = fmaxf(r.x, 0.f);
    r.y = fmaxf(r.y, 0.f);
    r.z = fmaxf(r.z, 0.f);
    r.w = fmaxf(r.w, 0.f);
  }
  *(v4f*)(out + (size_t)i * D + c * 4) = r;
}

extern "C" void kernel_launch(void* const* d_in, const int* in_sizes, int n_in,
                              void* d_out, int out_size, void* d_ws,
                              size_t ws_size, hipStream_t stream) {
  (void)in_sizes; (void)n_in; (void)out_size; (void)ws_size;

  const float* x  = (const float*)d_in[0];
  const int*   ei = (const int*)d_in[1];
  const int*   src = ei;            // edge_index[0]
  const int*   dst = ei + NE;       // edge_index[1]
  const float* W1 = (const float*)d_in[2];
  const float* b1 = (const float*)d_in[3];
  const float* W2 = (const float*)d_in[4];
  const float* b2 = (const float*)d_in[5];
  const float* W3 = (const float*)d_in[6];
  const float* b3 = (const float*)d_in[7];

  char* p = (char*)d_ws;
  auto take = [&](size_t bytes) -> void* {
    void* r = (void*)p;
    p += (bytes + 255) & ~(size_t)255;
    return r;
  };
  float* dinv = (float*)take((size_t)NN * 4);
  float* norm = (float*)take((size_t)NE * 4);
  float* bufA = (float*)take((size_t)NN * DH * 4);
  float* bufB = (float*)take((size_t)NN * DH * 4);
  float* bufC = (float*)take((size_t)NN * DH * 4);

  // ---- gcn_norm ----
  hipMemsetAsync(dinv, 0, (size_t)NN * 4, stream);
  k_degree<<<(NE + 255) / 256, 256, 0, stream>>>(dst, dinv, NE);
  k_dinv<<<(NN + 255) / 256, 256, 0, stream>>>(dinv, NN);
  k_norm<<<(NE + 255) / 256, 256, 0, stream>>>(src, dst, dinv, norm, NE);

  // ---- layer 1: x -> bufB (relu) ----
  k_gemm_wmma<DH / 16><<<NN / 16, (DH / 16) * 32, 0, stream>>>(x, W1, bufA);
  hipMemsetAsync(bufB, 0, (size_t)NN * DH * 4, stream);
  k_scatter<DH><<<(NE * (DH / 4) + 255) / 256, 256, 0, stream>>>(
      bufA, src, dst, norm, bufB, NE);
  k_finish<DH, true><<<(NN * (DH / 4) + 255) / 256, 256, 0, stream>>>(
      bufB, bufA, dinv, b1, bufB, NN);

  // ---- layer 2: bufB -> bufC (relu) ----
  k_gemm_wmma<DH / 16><<<NN / 16, (DH / 16) * 32, 0, stream>>>(bufB, W2, bufA);
  hipMemsetAsync(bufC, 0, (size_t)NN * DH * 4, stream);
  k_scatter<DH><<<(NE * (DH / 4) + 255) / 256, 256, 0, stream>>>(
      bufA, src, dst, norm, bufC, NE);
  k_finish<DH, true><<<(NN * (DH / 4) + 255) / 256, 256, 0, stream>>>(
      bufC, bufA, dinv, b2, bufC, NN);

  // ---- layer 3: bufC -> d_out (no relu) ----
  k_gemm_wmma<DO / 16><<<NN / 16, (DO / 16) * 32, 0, stream>>>(bufC, W3, bufA);
  hipMemsetAsync(bufB, 0, (size_t)NN * DO * 4, stream);
  k_scatter<DO><<<(NE * (DO / 4) + 255) / 256, 256, 0, stream>>>(
      bufA, src, dst, norm, bufB, NE);
  k_finish<DO, false><<<(NN * (DO / 4) + 255) / 256, 256, 0, stream>>>(
      bufB, bufA, dinv, b3, (float*)d_out, NN);
}